// PanopticGTGenerate_841813590616
// MI455X (gfx1250) — compile-verified
//
#include <hip/hip_runtime.h>
#include <hip/hip_bf16.h>
#include <stdint.h>

#define N_ROISC 1000
#define N_GTC 64
#define HWPIX (1024*1024)
#define NUM_CLS 81
#define THING_OFF 53   // NUM_SEG_CLASSES(133) - NUM_INST_CLASSES(80)
#define STUFF_MAXC 52  // NUM_SEG_CLASSES(133) - NUM_CLASSES(81)

typedef int   v4i __attribute__((ext_vector_type(4)));
typedef float v4f __attribute__((ext_vector_type(4)));

// ---- ws layout (bytes) ----
// [0,512):      u64 wkey[64]    packed (ovmax_bits<<32 | roi_idx), 0 = empty
// [512,4512):   int jmax[1000]
// [4512,8512):  int flags[1000] bit0=cand(ov>0.5), bit1=is_bg
// [8512,8768):  int vtab[64]    compacted value per GT (valid only where mask bit set)
// [8768,8776):  u32 vmask[2]    64-bit valid-GT mask

__global__ void k_init(unsigned long long* wkey) {
    wkey[threadIdx.x] = 0ull;
}

__global__ __launch_bounds__(256) void k_match(
        const float* __restrict__ rois, const float* __restrict__ cls,
        const int* __restrict__ label, const float* __restrict__ gt,
        unsigned long long* __restrict__ wkey,
        int* __restrict__ jmax, int* __restrict__ flags) {
    __shared__ float sg[N_GTC][4];
    int lt = threadIdx.x;
    if (lt < N_GTC) {
        sg[lt][0] = gt[lt*5+1]; sg[lt][1] = gt[lt*5+2];
        sg[lt][2] = gt[lt*5+3]; sg[lt][3] = gt[lt*5+4];
    }
    __syncthreads();
    int t = blockIdx.x * 256 + lt;
    if (t >= N_ROISC) return;

    float ax1 = rois[t*5+1], ay1 = rois[t*5+2], ax2 = rois[t*5+3], ay2 = rois[t*5+4];
    float areaA = (ax2-ax1+1.f)*(ay2-ay1+1.f);
    float best = -1.f; int bj = 0;
    for (int j = 0; j < N_GTC; ++j) {
        float bx1 = sg[j][0], by1 = sg[j][1], bx2 = sg[j][2], by2 = sg[j][3];
        float areaB = (bx2-bx1+1.f)*(by2-by1+1.f);
        float iw = fmaxf(fminf(ax2,bx2) - fmaxf(ax1,bx1) + 1.f, 0.f);
        float ih = fmaxf(fminf(ay2,by2) - fmaxf(ay1,by1) + 1.f, 0.f);
        float inter = iw*ih;
        float iou = inter / (areaA + areaB - inter);
        if (iou > best) { best = iou; bj = j; }   // first-max == jnp.argmax
    }
    float cb = cls[t*NUM_CLS]; int cj = 0;
    for (int c = 1; c < NUM_CLS; ++c) {
        float v = cls[t*NUM_CLS + c];
        if (v > cb) { cb = v; cj = c; }
    }
    int bg   = (cj == 0) && (label[t] == 0);
    int cand = best > 0.5f;
    jmax[t]  = bj;
    flags[t] = cand | (bg << 1);
    if (cand) {
        // positive floats compare as their uint bits; tie -> larger index wins,
        // matching the serial scan's "later equal-max evicts earlier" rule.
        unsigned long long key =
            ((unsigned long long)__float_as_uint(best) << 32) | (unsigned)t;
        atomicMax(&wkey[bj], key);
    }
}

__global__ __launch_bounds__(1024) void k_compact(
        const float* __restrict__ rois, const int* __restrict__ label,
        const unsigned long long* __restrict__ wkey,
        const int* __restrict__ jmax, const int* __restrict__ flags,
        float* __restrict__ out, int* __restrict__ vtab,
        unsigned* __restrict__ vmask) {
    __shared__ int s_scan[1024];
    __shared__ int s_keep[1024];
    __shared__ int s_mo[N_GTC];
    __shared__ unsigned s_vm[2];
    int i = threadIdx.x;
    if (i < N_GTC) {
        unsigned long long k = wkey[i];
        s_mo[i] = (k != 0ull) ? (int)(unsigned)(k & 0xffffffffull) : -1;
    }
    if (i < 2) s_vm[i] = 0u;
    if (i < N_ROISC) {  // zero rois_out + label_out regions (d_out is poisoned)
        out[i*5+0]=0.f; out[i*5+1]=0.f; out[i*5+2]=0.f; out[i*5+3]=0.f; out[i*5+4]=0.f;
        out[5000 + i] = 0.f;
    }
    __syncthreads();
    int keep = 0;
    if (i < N_ROISC) {
        int f = flags[i];
        keep = (f & 1) ? (s_mo[jmax[i]] == i) : !((f >> 1) & 1);
    }
    s_keep[i] = keep;
    s_scan[i] = keep;
    __syncthreads();
    for (int off = 1; off < 1024; off <<= 1) {   // inclusive Hillis-Steele
        int t = (i >= off) ? s_scan[i - off] : 0;
        __syncthreads();
        s_scan[i] += t;
        __syncthreads();
    }
    if (i < N_ROISC && s_keep[i]) {
        int dst = s_scan[i] - 1;
        out[dst*5+0] = 0.f;
        out[dst*5+1] = rois[i*5+1];
        out[dst*5+2] = rois[i*5+2];
        out[dst*5+3] = rois[i*5+3];
        out[dst*5+4] = rois[i*5+4];
        out[5000 + dst] = (float)label[i];
    }
    if (i < N_GTC) {
        int mo = s_mo[i];
        int vv = 0;
        if (mo >= 0) {
            vv = s_scan[mo] - 1 + THING_OFF;
            atomicOr(&s_vm[i >> 5], 1u << (i & 31));
        }
        vtab[i] = vv;
    }
    __syncthreads();
    if (i < 2) vmask[i] = s_vm[i];
    if (i == 0) out[6000 + HWPIX] = (float)s_scan[N_ROISC - 1];   // num_keep
}

// The hot kernel: 264 MB single-pass stream -> ~11.3 us HBM floor.
// 4 px/thread, 64 coalesced b128 NT loads/thread (max MLP under LOADcnt),
// seg_gt tile prefetched via CDNA5 async global->LDS op, overlapped with
// the 64-GT mask stream; consumed after s_wait_asynccnt 0 + barrier.
__global__ __launch_bounds__(256) void k_mg(
        const int* __restrict__ seg, const int* __restrict__ mask,
        const int* __restrict__ vtab, const unsigned* __restrict__ vmask,
        float* __restrict__ mg) {
    __shared__ int s_seg[1024];
    __shared__ int s_v[N_GTC];
    __shared__ unsigned s_vm[2];
    int tid = threadIdx.x;
    int p4 = blockIdx.x * 1024 + tid * 4;

    // Async tensor-path copy: 16B/lane, 4KB/block of seg_gt into LDS.
    // LDS byte address = low 32 bits of the generic pointer (ISA 10.2);
    // vdst carries the LDS address.
    {
        unsigned lds_dst = (unsigned)(uintptr_t)(void*)&s_seg[tid * 4];
        const int* gsrc = seg + p4;
        asm volatile("global_load_async_to_lds_b128 %0, %1, off"
                     :: "v"(lds_dst), "v"(gsrc)
                     : "memory");
    }
    if (tid < N_GTC) s_v[tid] = vtab[tid];
    if (tid < 2)     s_vm[tid] = vmask[tid];

    unsigned lo[4] = {0,0,0,0};
    unsigned hi[4] = {0,0,0,0};
    const v4i* mp = (const v4i*)mask + (p4 >> 2);
#pragma unroll
    for (int i = 0; i < 64; ++i) {
        v4i m = __builtin_nontemporal_load(mp + (size_t)i * (HWPIX / 4));
        unsigned b = 1u << (i & 31);
        if (i < 32) {
            if (m.x) lo[0] |= b; if (m.y) lo[1] |= b;
            if (m.z) lo[2] |= b; if (m.w) lo[3] |= b;
        } else {
            if (m.x) hi[0] |= b; if (m.y) hi[1] |= b;
            if (m.z) hi[2] |= b; if (m.w) hi[3] |= b;
        }
    }
    asm volatile("s_wait_asynccnt 0" ::: "memory");
    __syncthreads();
    unsigned long long vm = ((unsigned long long)s_vm[1] << 32) | s_vm[0];
    v4f r;
#pragma unroll
    for (int k = 0; k < 4; ++k) {
        unsigned long long bm = (((unsigned long long)hi[k] << 32) | lo[k]) & vm;
        int idx = 63 - __builtin_clzll(bm | 1ull);   // |1 avoids clz(0) poison; unused when bm==0
        float fv = (float)s_v[idx];
        int s = s_seg[tid * 4 + k];
        float fb = (s <= STUFF_MAXC || s >= 255) ? (float)s : 255.0f;
        r[k] = bm ? fv : fb;
    }
    __builtin_nontemporal_store(r, (v4f*)(mg + p4));
}

extern "C" void kernel_launch(void* const* d_in, const int* in_sizes, int n_in,
                              void* d_out, int out_size, void* d_ws, size_t ws_size,
                              hipStream_t stream) {
    const float* rois  = (const float*)d_in[0];
    // d_in[1] bbox_pred: unused
    const float* cls   = (const float*)d_in[2];
    const int*   label = (const int*)d_in[3];
    const float* gt    = (const float*)d_in[4];
    // d_in[5] cls_idx: unused
    const int*   seg   = (const int*)d_in[6];
    const int*   mask  = (const int*)d_in[7];
    // d_in[8] im_shape: unused
    float* out = (float*)d_out;

    char* ws = (char*)d_ws;
    unsigned long long* wkey = (unsigned long long*)(ws);
    int*      jmax  = (int*)(ws + 512);
    int*      flags = (int*)(ws + 4512);
    int*      vtab  = (int*)(ws + 8512);
    unsigned* vmask = (unsigned*)(ws + 8768);

    k_init<<<1, 64, 0, stream>>>(wkey);
    k_match<<<(N_ROISC + 255) / 256, 256, 0, stream>>>(rois, cls, label, gt,
                                                       wkey, jmax, flags);
    k_compact<<<1, 1024, 0, stream>>>(rois, label, wkey, jmax, flags,
                                      out, vtab, vmask);
    k_mg<<<HWPIX / 1024, 256, 0, stream>>>(seg, mask, vtab, vmask, out + 6000);
}